// PositionSematicLoss_53661321396388
// MI455X (gfx1250) — compile-verified
//
#include <hip/hip_runtime.h>
#include <hip/hip_bf16.h>

typedef __attribute__((ext_vector_type(2))) float v2f;
typedef __attribute__((ext_vector_type(4))) float v4f;
typedef __attribute__((ext_vector_type(8))) float v8f;

#define BATCH 64
#define DDIM (64 * 64 * 64)      // 262144
#define NWG 256                  // workgroups in main kernel
#define KRANGE (DDIM / NWG)      // 1024 K per workgroup
#define KS 32                    // LDS tile width (K per staging step)
#define LSTR 36                  // padded LDS row stride in floats (144B: 16B-aligned, conflict-free b64 reads)
#define NITER (KRANGE / KS)      // 32 double-buffered steps

// ws layout (floats):
//   [0      .. 4095 ]  gram P·P^T   (64x64)
//   [4096   .. 8191 ]  gram S·S^T
//   [8192   .. 12287]  gram O·O^T
//   [12288  .. 16383]  cross S·O^T
//   [16384  .. 16447]  rowdot diag(P·O^T)
#define WS_FLOATS (4 * 4096 + 64)

__global__ __launch_bounds__(256) void zero_ws_kernel(float* __restrict__ w, int n) {
    int i = blockIdx.x * blockDim.x + threadIdx.x;
    if (i < n) w[i] = 0.0f;
}

__global__ __launch_bounds__(256) void gram_fused_kernel(
    const float* __restrict__ P, const float* __restrict__ S,
    const float* __restrict__ O, float* __restrict__ gram,
    float* __restrict__ rowdot) {
    __shared__ float lds[2][3][BATCH][LSTR];   // 55,296 bytes

    const int tid  = threadIdx.x;
    const int wave = tid >> 5;
    const int lane = tid & 31;
    const int kbeg = blockIdx.x * KRANGE;

    // wave role: product 0=P·P^T 1=S·S^T 2=O·O^T 3=S·O^T ; rh = row half
    const int prod = wave >> 1;
    const int rh   = wave & 1;
    const int matA = (prod == 3) ? 1 : prod;
    const int matB = (prod == 3) ? 2 : prod;

    // WMMA fragment addressing (f32 16x16x4): lane L -> m/n = L%16 ; K pair base = (L/16)*2
    const int m16   = lane & 15;
    const int khalf = (lane >> 4) << 1;

    v8f acc[2][4];
#pragma unroll
    for (int j = 0; j < 2; ++j)
#pragma unroll
        for (int c = 0; c < 4; ++c)
#pragma unroll
            for (int v = 0; v < 8; ++v) acc[j][c][v] = 0.0f;

    // staging + rowdot role: 4 threads per row, 8 consecutive floats each
    const int srow = tid >> 2;
    const int scol = (tid & 3) << 3;    // 0,8,16,24
    float rd_acc = 0.0f;

    // Issue one 64xKS tile for all three matrices: 6 async b128 loads per thread.
    // GVS addressing: 64-bit scalar base + 32-bit per-lane byte offset; LDS dest
    // address = low 32 bits of the generic __shared__ pointer.
    auto issue_tile = [&](int buf, int kt) {
        const unsigned goff = (unsigned)(srow * DDIM + kbeg + kt + scol) * 4u;
        const unsigned l0 = (unsigned)(uintptr_t)&lds[buf][0][srow][scol];
        const unsigned l1 = (unsigned)(uintptr_t)&lds[buf][1][srow][scol];
        const unsigned l2 = (unsigned)(uintptr_t)&lds[buf][2][srow][scol];
        asm volatile(
            "global_load_async_to_lds_b128 %0, %3, %4 offset:0\n\t"
            "global_load_async_to_lds_b128 %0, %3, %4 offset:16\n\t"
            "global_load_async_to_lds_b128 %1, %3, %5 offset:0\n\t"
            "global_load_async_to_lds_b128 %1, %3, %5 offset:16\n\t"
            "global_load_async_to_lds_b128 %2, %3, %6 offset:0\n\t"
            "global_load_async_to_lds_b128 %2, %3, %6 offset:16"
            :: "v"(l0), "v"(l1), "v"(l2), "v"(goff), "s"(P), "s"(S), "s"(O)
            : "memory");
    };

    issue_tile(0, 0);
    issue_tile(1, KS);

    for (int i = 0; i < NITER; ++i) {
        const int buf = i & 1;
        // current buffer's 6 loads are the oldest outstanding (in-order completion)
        if (i + 1 < NITER) asm volatile("s_wait_asynccnt 6" ::: "memory");
        else               asm volatile("s_wait_asynccnt 0" ::: "memory");
        __syncthreads();

        // rowdot partial: P elementwise O
#pragma unroll
        for (int q = 0; q < 8; ++q)
            rd_acc += lds[buf][0][srow][scol + q] * lds[buf][2][srow][scol + q];

        // WMMA sweep over this LDS tile
#pragma unroll
        for (int k0 = 0; k0 < KS; k0 += 4) {
            v2f fa[2], fb[4];
#pragma unroll
            for (int j = 0; j < 2; ++j)
                fa[j] = *(const v2f*)&lds[buf][matA][rh * 32 + j * 16 + m16][k0 + khalf];
#pragma unroll
            for (int c = 0; c < 4; ++c)
                fb[c] = *(const v2f*)&lds[buf][matB][c * 16 + m16][k0 + khalf];
#pragma unroll
            for (int j = 0; j < 2; ++j)
#pragma unroll
                for (int c = 0; c < 4; ++c)
                    acc[j][c] = __builtin_amdgcn_wmma_f32_16x16x4_f32(
                        false, fa[j], false, fb[c], (short)0, acc[j][c], false, false);
        }
        __syncthreads();   // all waves done with lds[buf] before it is overwritten

        if (i + 2 < NITER) issue_tile(buf, (i + 2) * KS);
    }

    // accumulate into global workspace
    float* g = gram + prod * (BATCH * BATCH);
    const int mbase = (lane >> 4) << 3;  // C/D layout: lanes 16-31 hold M = v+8
#pragma unroll
    for (int j = 0; j < 2; ++j) {
        const int rowblock = (rh * 2 + j) * 16;
#pragma unroll
        for (int c = 0; c < 4; ++c) {
            const int colblock = c * 16;
#pragma unroll
            for (int v = 0; v < 8; ++v)
                atomicAdd(&g[(rowblock + mbase + v) * BATCH + colblock + m16], acc[j][c][v]);
        }
    }
    atomicAdd(&rowdot[srow], rd_acc);
}

__global__ __launch_bounds__(64) void finalize_kernel(const float* __restrict__ ws,
                                                      float* __restrict__ out) {
    __shared__ float np[BATCH], ns[BATCH], no[BATCH];
    __shared__ float r0[BATCH], r1[BATCH], r2[BATCH], r3[BATCH], r4[BATCH];
    const int b = threadIdx.x;
    const float* gP  = ws;
    const float* gS  = ws + 4096;
    const float* gO  = ws + 8192;
    const float* gSO = ws + 12288;
    const float* rd  = ws + 16384;
    const float eps = 1e-8f;

    np[b] = fmaxf(sqrtf(fmaxf(gP[b * BATCH + b], 0.0f)), eps);
    ns[b] = fmaxf(sqrtf(fmaxf(gS[b * BATCH + b], 0.0f)), eps);
    no[b] = fmaxf(sqrtf(fmaxf(gO[b * BATCH + b], 0.0f)), eps);
    __syncthreads();

    float sp = 0.0f, ss = 0.0f, sgo = 0.0f, srow = 0.0f;
    for (int j = 0; j < BATCH; ++j) {
        float p = gP[b * BATCH + j] / (np[b] * np[j]);
        float s = gS[b * BATCH + j] / (ns[b] * ns[j]);
        if (j != b) { sp += p; ss += s; }
        sgo  += gO[b * BATCH + j];
        srow += gSO[b * BATCH + j];
    }
    r0[b] = sp;
    r1[b] = ss;
    r2[b] = sgo;                                   // total sum of G_orig
    r3[b] = rd[b] / (np[b] * no[b]);               // sim_1[b]
    r4[b] = srow / ((float)BATCH * ns[b]);         // sim_2[b] * ||om||
    __syncthreads();

    if (b == 0) {
        float R0 = 0, R1 = 0, R2 = 0, R3 = 0, R4 = 0;
        for (int j = 0; j < BATCH; ++j) {
            R0 += r0[j]; R1 += r1[j]; R2 += r2[j]; R3 += r3[j]; R4 += r4[j];
        }
        const float denom = (float)BATCH * (float)(BATCH - 1);
        float pos_loss = R0 / denom;
        float sem_loss = R1 / denom;
        float INN_loss = 0.2f * pos_loss - sem_loss;

        float om_norm = fmaxf(sqrtf(fmaxf(R2, 0.0f)) / (float)BATCH, eps);
        float sim_loss_1 = -R3 / (float)BATCH;
        float sim_loss_2 = -(R4 / om_norm) / (float)BATCH;
        out[0] = INN_loss;
        out[1] = 0.2f * sim_loss_1 + sim_loss_2;
    }
}

extern "C" void kernel_launch(void* const* d_in, const int* in_sizes, int n_in,
                              void* d_out, int out_size, void* d_ws, size_t ws_size,
                              hipStream_t stream) {
    const float* P = (const float*)d_in[0];
    const float* S = (const float*)d_in[1];
    const float* O = (const float*)d_in[2];
    float* ws  = (float*)d_ws;
    float* out = (float*)d_out;

    zero_ws_kernel<<<(WS_FLOATS + 255) / 256, 256, 0, stream>>>(ws, WS_FLOATS);
    gram_fused_kernel<<<NWG, 256, 0, stream>>>(P, S, O, ws, ws + 4 * 4096);
    finalize_kernel<<<1, 64, 0, stream>>>(ws, out);
}